// Regression_43078521979263
// MI455X (gfx1250) — compile-verified
//
#include <hip/hip_runtime.h>
#include <stdint.h>

// ---------------------------------------------------------------------------
// 11x11 bandwidth-weighted sliding-window regression (bilateral-style).
// VALU/trans-bound; halo tiles are DMA'd to LDS by the CDNA5 Tensor Data
// Mover (tensor_load_to_lds / TENSORcnt). Interior tiles process TWO pixels
// per lane on VOP3P packed f32 (inline-asm v_pk_add/mul/fma_f32), fed from a
// pair-interleaved LDS layout so each ds_load_b64 lands an {A,B} operand
// directly in an aligned VGPR pair (no shuffles). v_exp_f32 stays scalar.
// ---------------------------------------------------------------------------

#define TILE     32
#define RWIN     5
#define WIN      11
#define HALO     (TILE + 2 * RWIN)   // 42
#define PITCH    (3 * HALO)          // 126 floats per LDS row (odd dword pixel stride)
#define LDSN     (HALO * PITCH)      // 5292 floats per raw buffer
#define NTHREADS 256
#define ROWOFF   (16 * PITCH)        // raw-layout offset between paired pixels
#define PAIRROWS 26                  // A-partner rows: lyA+dy in 0..25
#define PAIRN    (PAIRROWS * PITCH)  // 3276 v2f entries per pair buffer

typedef float        v2f       __attribute__((ext_vector_type(2)));
typedef unsigned int tdm_u32x4 __attribute__((ext_vector_type(4)));
typedef int          tdm_i32x8 __attribute__((ext_vector_type(8)));
typedef int          tdm_i32x4 __attribute__((ext_vector_type(4)));

#if defined(__gfx1250__) && __has_builtin(__builtin_amdgcn_tensor_load_to_lds)
#define USE_TDM 1
#else
#define USE_TDM 0
#endif

// ---- VOP3P packed-f32 helpers (guaranteed v_pk_* emission) ----------------
#if defined(__gfx1250__)
__device__ __forceinline__ v2f pk_sub(v2f a, v2f b) {        // a - b
    v2f d;
    asm("v_pk_add_f32 %0, %1, %2 neg_lo:[0,1] neg_hi:[0,1]"
        : "=v"(d) : "v"(a), "v"(b));
    return d;
}
__device__ __forceinline__ v2f pk_add(v2f a, v2f b) {
    v2f d;
    asm("v_pk_add_f32 %0, %1, %2" : "=v"(d) : "v"(a), "v"(b));
    return d;
}
__device__ __forceinline__ v2f pk_mul(v2f a, v2f b) {
    v2f d;
    asm("v_pk_mul_f32 %0, %1, %2" : "=v"(d) : "v"(a), "v"(b));
    return d;
}
__device__ __forceinline__ v2f pk_fma(v2f a, v2f b, v2f c) { // a*b + c
    v2f d;
    asm("v_pk_fma_f32 %0, %1, %2, %3" : "=v"(d) : "v"(a), "v"(b), "v"(c));
    return d;
}
#else
__device__ __forceinline__ v2f pk_sub(v2f a, v2f b) { return a - b; }
__device__ __forceinline__ v2f pk_add(v2f a, v2f b) { return a + b; }
__device__ __forceinline__ v2f pk_mul(v2f a, v2f b) { return a * b; }
__device__ __forceinline__ v2f pk_fma(v2f a, v2f b, v2f c) { return a * b + c; }
#endif

#if USE_TDM
// Build a D# (CDNA5 ISA ch.10.8) for a 2D tile of f32 rows:
//   group0: count=1 | lds_addr | global_addr(57b) | type=2
//   group1: data_size=4B, tensor_dim0/1 = clamped tile dims, tile_dim0 =
//           dwords/row, tile_dim1 = rows, tensor_dim0_stride = row stride.
__device__ __forceinline__ void tdm_load_tile_2d(unsigned lds_byte_addr,
                                                 const void* gptr,
                                                 unsigned w_elems,
                                                 unsigned h_rows,
                                                 unsigned row_stride_elems)
{
    unsigned long long ga = (unsigned long long)(uintptr_t)gptr;
    tdm_u32x4 g0;
    g0.x = 1u;                                                   // count=1, user D#
    g0.y = lds_byte_addr;                                        // lds_addr [63:32]
    g0.z = (unsigned)ga;                                         // global_addr lo
    g0.w = (unsigned)((ga >> 32) & 0x01FFFFFFull) | 0x80000000u; // addr hi | type=2

    tdm_i32x8 g1;
    g1.s0 = 0x00020000;                                          // data_size=2 (4B)
    g1.s1 = (int)((w_elems & 0xFFFFu) << 16);                    // tensor_dim0[15:0]
    g1.s2 = (int)(((w_elems >> 16) & 0xFFFFu) |                  // tensor_dim0[31:16]
                  ((h_rows & 0xFFFFu) << 16));                   // tensor_dim1[15:0]
    g1.s3 = (int)(((h_rows >> 16) & 0xFFFFu) |                   // tensor_dim1[31:16]
                  ((w_elems & 0xFFFFu) << 16));                  // tile_dim0
    g1.s4 = (int)(h_rows & 0xFFFFu);                             // tile_dim1 (tile_dim2=0)
    g1.s5 = (int)row_stride_elems;                               // tensor_dim0_stride lo32
    g1.s6 = 0;
    g1.s7 = 0;

    tdm_i32x4 z4 = {0, 0, 0, 0};
#if __has_include(<hip/amd_detail/amd_gfx1250_TDM.h>)
    tdm_i32x8 z8 = {0, 0, 0, 0, 0, 0, 0, 0};
    __builtin_amdgcn_tensor_load_to_lds(g0, g1, z4, z4, z8, 0);  // clang-23 6-arg form
#else
    __builtin_amdgcn_tensor_load_to_lds(g0, g1, z4, z4, 0);      // ROCm 7.2 5-arg form
#endif
}
#endif

__global__ __launch_bounds__(NTHREADS) void regression_tile_kernel(
    const float* __restrict__ rnd, const float* __restrict__ tgt,
    float* __restrict__ out, int B, int H, int W, size_t N)
{
    __shared__ float t_s[LDSN];      // raw tile (TDM destination; edge path)
    __shared__ float r_s[LDSN];
    __shared__ v2f   tp_v[PAIRN];    // pair-interleaved {row r, row r+16}
    __shared__ v2f   rp_v[PAIRN];

    const int b   = blockIdx.z;
    const int ty0 = blockIdx.y * TILE;
    const int tx0 = blockIdx.x * TILE;
    const int tid = threadIdx.x;

    // clamped halo rectangle of this tile
    const int hy0 = ty0 - RWIN, hx0 = tx0 - RWIN;
    const int yl = hy0 < 0 ? 0 : hy0;
    const int xl = hx0 < 0 ? 0 : hx0;
    int yh = hy0 + HALO; if (yh > H) yh = H;
    int xh = hx0 + HALO; if (xh > W) xh = W;
    const int hrows = yh - yl;
    const int wpix  = xh - xl;
    const bool interior = (hrows == HALO) && (wpix == HALO);   // block-uniform

#if USE_TDM
    // zero-fill only border tiles (keeps 0*garbage / NaN out of accumulators;
    // weights there are also forced to 0 in the checked loop)
    if (!interior) {
        for (int i = tid; i < LDSN; i += NTHREADS) { t_s[i] = 0.0f; r_s[i] = 0.0f; }
    }
    __syncthreads();
    if (tid < 32) {                              // wave 0 only (uniform branch)
        const unsigned ldsoff =
            (unsigned)(((yl - hy0) * PITCH + 3 * (xl - hx0)) * 4);
        const size_t gelem = ((size_t)(b * H + yl) * W + xl) * 3;
        tdm_load_tile_2d((unsigned)(uintptr_t)t_s + ldsoff, tgt + gelem,
                         (unsigned)(3 * wpix), (unsigned)hrows, (unsigned)(3 * W));
        tdm_load_tile_2d((unsigned)(uintptr_t)r_s + ldsoff, rnd + gelem,
                         (unsigned)(3 * wpix), (unsigned)hrows, (unsigned)(3 * W));
        __builtin_amdgcn_s_wait_tensorcnt(0);
    }
    __syncthreads();
#else
    // cooperative fallback fill (no TDM builtin available)
    for (int i = tid; i < HALO * HALO; i += NTHREADS) {
        const int ry = i / HALO, rx = i - ry * HALO;
        const int gy = hy0 + ry, gx = hx0 + rx;
        const int l  = ry * PITCH + 3 * rx;
        if ((unsigned)gy < (unsigned)H && (unsigned)gx < (unsigned)W) {
            const size_t g = ((size_t)(b * H + gy) * W + gx) * 3;
            t_s[l] = tgt[g]; t_s[l + 1] = tgt[g + 1]; t_s[l + 2] = tgt[g + 2];
            r_s[l] = rnd[g]; r_s[l + 1] = rnd[g + 1]; r_s[l + 2] = rnd[g + 2];
        } else {
            t_s[l] = 0.f; t_s[l + 1] = 0.f; t_s[l + 2] = 0.f;
            r_s[l] = 0.f; r_s[l + 1] = 0.f; r_s[l + 2] = 0.f;
        }
    }
    __syncthreads();
#endif

    // exp(-d/bw^2) = exp2(d * c2), c2 = -log2(e)/0.25 = -4*log2(e)
    const float c2 = -5.7707801635558535f;

    if (interior) {
        // pair-interleave build: tp_v[i] = { raw[i], raw[i + 16 rows] }
        // (raw pitch == pair pitch == 126 floats, so indices are identical)
        for (int i = tid; i < PAIRN; i += NTHREADS) {
            tp_v[i] = (v2f){ t_s[i], t_s[i + ROWOFF] };
            rp_v[i] = (v2f){ r_s[i], r_s[i + ROWOFF] };
        }
        __syncthreads();

        const v2f c2v = { c2, c2 };
        const int lx  = tid & 31;
        const int gx  = tx0 + lx;
#pragma unroll 1
        for (int pair = 0; pair < 2; ++pair) {
            const int lyA = (tid >> 5) + pair * 8;     // 0..15
            const int cb  = (lyA + RWIN) * PITCH + 3 * (lx + RWIN);
            const v2f tc0 = tp_v[cb + 0];
            const v2f tc1 = tp_v[cb + 1];
            const v2f tc2 = tp_v[cb + 2];

            v2f sx0 = {0.f, 0.f}, sx1 = {0.f, 0.f}, sx2 = {0.f, 0.f};
            v2f sxx0 = {0.f, 0.f}, sxx1 = {0.f, 0.f}, sxx2 = {0.f, 0.f};
            v2f sxy0 = {0.f, 0.f}, sxy1 = {0.f, 0.f}, sxy2 = {0.f, 0.f};
            v2f sy0 = {0.f, 0.f}, sy1 = {0.f, 0.f}, sy2 = {0.f, 0.f};
            v2f swt = {0.f, 0.f};

            for (int dy = 0; dy < WIN; ++dy) {
                const int rb = (lyA + dy) * PITCH + 3 * lx;
#pragma unroll
                for (int dx = 0; dx < WIN; ++dx) {
                    const int pb = rb + 3 * dx;
                    const v2f t0 = tp_v[pb + 0];
                    const v2f t1 = tp_v[pb + 1];
                    const v2f t2 = tp_v[pb + 2];
                    const v2f r0 = rp_v[pb + 0];
                    const v2f r1 = rp_v[pb + 1];
                    const v2f r2 = rp_v[pb + 2];

                    const v2f d0 = pk_sub(tc0, t0);
                    const v2f d1 = pk_sub(tc1, t1);
                    const v2f d2 = pk_sub(tc2, t2);
                    v2f dist = pk_mul(d0, d0);
                    dist = pk_fma(d1, d1, dist);
                    dist = pk_fma(d2, d2, dist);
                    const v2f ea = pk_mul(dist, c2v);
                    const v2f w  = { __builtin_amdgcn_exp2f(ea.x),
                                     __builtin_amdgcn_exp2f(ea.y) };

                    swt = pk_add(swt, w);
                    const v2f wt0 = pk_mul(w, t0);
                    const v2f wt1 = pk_mul(w, t1);
                    const v2f wt2 = pk_mul(w, t2);
                    sx0  = pk_add(sx0, wt0);
                    sx1  = pk_add(sx1, wt1);
                    sx2  = pk_add(sx2, wt2);
                    sxx0 = pk_fma(wt0, t0, sxx0);
                    sxx1 = pk_fma(wt1, t1, sxx1);
                    sxx2 = pk_fma(wt2, t2, sxx2);
                    sxy0 = pk_fma(wt0, r0, sxy0);
                    sxy1 = pk_fma(wt1, r1, sxy1);
                    sxy2 = pk_fma(wt2, r2, sxy2);
                    sy0  = pk_fma(w, r0, sy0);
                    sy1  = pk_fma(w, r1, sy1);
                    sy2  = pk_fma(w, r2, sy2);
                }
            }

            const float invA = __builtin_amdgcn_rcpf(swt.x);
            const float invB = __builtin_amdgcn_rcpf(swt.y);
            const int gyA = ty0 + lyA, gyB = gyA + 16;
            const size_t nA = ((size_t)(b * H + gyA) * W + gx) * 3;
            const size_t nB = ((size_t)(b * H + gyB) * W + gx) * 3;
            out[nA + 0] = sx0.x * invA;  out[nA + 1] = sx1.x * invA;  out[nA + 2] = sx2.x * invA;
            out[nB + 0] = sx0.y * invB;  out[nB + 1] = sx1.y * invB;  out[nB + 2] = sx2.y * invB;
            out[N + nA + 0] = sxx0.x * invA; out[N + nA + 1] = sxx1.x * invA; out[N + nA + 2] = sxx2.x * invA;
            out[N + nB + 0] = sxx0.y * invB; out[N + nB + 1] = sxx1.y * invB; out[N + nB + 2] = sxx2.y * invB;
            out[2 * N + nA + 0] = sxy0.x * invA; out[2 * N + nA + 1] = sxy1.x * invA; out[2 * N + nA + 2] = sxy2.x * invA;
            out[2 * N + nB + 0] = sxy0.y * invB; out[2 * N + nB + 1] = sxy1.y * invB; out[2 * N + nB + 2] = sxy2.y * invB;
            out[3 * N + nA + 0] = sy0.x * invA;  out[3 * N + nA + 1] = sy1.x * invA;  out[3 * N + nA + 2] = sy2.x * invA;
            out[3 * N + nB + 0] = sy0.y * invB;  out[3 * N + nB + 1] = sy1.y * invB;  out[3 * N + nB + 2] = sy2.y * invB;
        }
    } else {
        // ----- border tiles (~13% of blocks): checked scalar loop ---------
        for (int px = tid; px < TILE * TILE; px += NTHREADS) {
            const int ly = px >> 5, lx = px & 31;
            const int gy = ty0 + ly, gx = tx0 + lx;
            if (gy >= H) continue;
            const int cbase = (ly + RWIN) * PITCH + 3 * (lx + RWIN);
            const float tc0 = t_s[cbase], tc1 = t_s[cbase + 1], tc2 = t_s[cbase + 2];
            float sx0 = 0.f, sx1 = 0.f, sx2 = 0.f;
            float sxx0 = 0.f, sxx1 = 0.f, sxx2 = 0.f;
            float sxy0 = 0.f, sxy1 = 0.f, sxy2 = 0.f;
            float sy0 = 0.f, sy1 = 0.f, sy2 = 0.f, swt = 0.f;
            for (int dy = 0; dy < WIN; ++dy) {
                const bool rowOk = (unsigned)(gy + dy - RWIN) < (unsigned)H;
                const int rbase = (ly + dy) * PITCH + 3 * lx;
#pragma unroll
                for (int dx = 0; dx < WIN; ++dx) {
                    const int idx = rbase + 3 * dx;
                    const float t0 = t_s[idx], t1 = t_s[idx + 1], t2 = t_s[idx + 2];
                    const float r0 = r_s[idx], r1 = r_s[idx + 1], r2 = r_s[idx + 2];
                    const float d0 = tc0 - t0, d1 = tc1 - t1, d2 = tc2 - t2;
                    const float dist = d0 * d0 + d1 * d1 + d2 * d2;
                    const bool ok = rowOk &&
                        ((unsigned)(gx + dx - RWIN) < (unsigned)W);
                    float w = __builtin_amdgcn_exp2f(dist * c2);
                    w = ok ? w : 0.0f;
                    swt += w;
                    const float wt0 = w * t0, wt1 = w * t1, wt2 = w * t2;
                    sx0  += wt0;        sx1  += wt1;        sx2  += wt2;
                    sxx0 += wt0 * t0;   sxx1 += wt1 * t1;   sxx2 += wt2 * t2;
                    sxy0 += wt0 * r0;   sxy1 += wt1 * r1;   sxy2 += wt2 * r2;
                    sy0  += w * r0;     sy1  += w * r1;     sy2  += w * r2;
                }
            }
            const float invw = __builtin_amdgcn_rcpf(swt);
            const size_t n0 = ((size_t)(b * H + gy) * W + gx) * 3;
            out[n0 + 0]         = sx0 * invw;
            out[n0 + 1]         = sx1 * invw;
            out[n0 + 2]         = sx2 * invw;
            out[N + n0 + 0]     = sxx0 * invw;
            out[N + n0 + 1]     = sxx1 * invw;
            out[N + n0 + 2]     = sxx2 * invw;
            out[2 * N + n0 + 0] = sxy0 * invw;
            out[2 * N + n0 + 1] = sxy1 * invw;
            out[2 * N + n0 + 2] = sxy2 * invw;
            out[3 * N + n0 + 0] = sy0 * invw;
            out[3 * N + n0 + 1] = sy1 * invw;
            out[3 * N + n0 + 2] = sy2 * invw;
        }
    }
}

extern "C" void kernel_launch(void* const* d_in, const int* in_sizes, int n_in,
                              void* d_out, int out_size, void* d_ws, size_t ws_size,
                              hipStream_t stream)
{
    (void)n_in; (void)out_size; (void)d_ws; (void)ws_size;
    const float* rnd = (const float*)d_in[0];   // rand
    const float* tgt = (const float*)d_in[1];   // target
    float* out = (float*)d_out;

    const int H = 720, W = 1280, C = 3;
    const int B = in_sizes[0] / (H * W * C);    // 4
    const size_t N = (size_t)B * H * W * C;

    dim3 grid((W + TILE - 1) / TILE, (H + TILE - 1) / TILE, B);
    dim3 block(NTHREADS);
    regression_tile_kernel<<<grid, block, 0, stream>>>(rnd, tgt, out, B, H, W, N);
}